// SelectiveScan_40535901340370
// MI455X (gfx1250) — compile-verified
//
#include <hip/hip_runtime.h>
#include <hip/hip_bf16.h>

// ---------------------------------------------------------------------------
// Mamba selective-scan block for MI455X (gfx1250), wave32, WMMA f32 path.
// B=1, L=2048, D_MODEL=1024, D_INNER=2048, D_STATE=16, D_CONV=4, DT_RANK=64
// GEMM shapes/strides are compile-time constants; each wave computes an
// MT*16 x NT*16 block so every B fragment feeds MT WMMAs from registers.
// A row-tiles addressed via constant immediate offsets from ONE base pointer
// (keeps loads on the GLOBAL path, not FLAT).
// ---------------------------------------------------------------------------

#define L_SEQ    2048
#define DMODEL   1024
#define DINNER   2048
#define DSTATE   16
#define DTRANK   64
#define XDBL_W   (DTRANK + 2 * DSTATE)   // 96

typedef __attribute__((ext_vector_type(2))) float v2f;
typedef __attribute__((ext_vector_type(8))) float v8f;

__device__ __forceinline__ float silu_f(float x) {
    return x * (1.0f / (1.0f + __expf(-x)));
}
__device__ __forceinline__ float softplus_f(float x) {
    return (x > 20.0f) ? x : log1pf(__expf(x));
}

// ---------------------------------------------------------------------------
// f32 WMMA GEMM:  C = epilogue( preop(A) @ B )
//   A: M x K (LDA), B: K x N (LDB), C: M x N (LDC) -- template constants.
//   PRE  == 1: a <- a * silu(A2[row, k])     (fused y * silu(z))
//   POST == 1: c <- softplus(c + bias[col])  (fused delta epilogue)
// One wave computes MT x NT tiles of 16x16: per K-step it loads MT A
// fragments (one global_load_b64 each, immediate row offsets) and NT B
// fragments, issuing MT*NT v_wmma_f32_16x16x4_f32 with full register reuse.
// Fragment layouts per CDNA5 ISA 7.12.2 (32-bit A 16x4 / 32-bit C 16x16):
//   A: lane half=0 -> K pair {k,k+1}, half=1 -> {k+2,k+3}; row = lane&15
//   B: same K-pair split; col = lane&15
//   C/D: VGPR r -> row (r + 8*half), col = lane&15
// ---------------------------------------------------------------------------
template <int PRE, int POST, int M, int N, int K,
          int LDA, int LDA2, int LDB, int LDC, int MT, int NT>
__global__ __launch_bounds__(128)
void gemm_wmma_f32(const float* __restrict__ A,
                   const float* __restrict__ A2,
                   const float* __restrict__ Bm,
                   const float* __restrict__ bias,
                   float* __restrict__ C)
{
    constexpr int tilesM  = M >> 4;
    constexpr int tilesN  = N >> 4;
    constexpr int groupsM = tilesM / MT;
    constexpr int groupsN = tilesN / NT;
    static_assert(tilesM % MT == 0 && tilesN % NT == 0, "exact tiling required");

    const int lane = threadIdx.x & 31;
    const int wave = threadIdx.x >> 5;
    const int half = lane >> 4;
    const int l    = lane & 15;

    const int wid = blockIdx.x * (blockDim.x >> 5) + wave;
    if (wid >= groupsM * groupsN) return;
    const int gm = wid / groupsN;
    const int gn = wid % groupsN;

    const int row0 = gm * (MT * 16) + l;
    const int col0 = gn * (NT * 16) + l;

    // Single base pointers; MT row-tiles reached via constant offsets
    // (m*16*LDA floats -> 24-bit immediate byte offsets in the loads).
    const float* __restrict__ Ap  = A + (size_t)row0 * LDA + half * 2;
    const float* __restrict__ A2p = PRE ? (A2 + (size_t)row0 * LDA2 + half * 2) : nullptr;
    const float* __restrict__ Bp  = Bm + (size_t)(half * 2) * LDB + col0;

    // one-shot streaming hint; everything is L2-resident afterwards
    __builtin_prefetch((const void*)Ap, 0, 1);          // global_prefetch_b8
    __builtin_prefetch((const void*)Bp, 0, 1);

    v8f acc[MT][NT];
    #pragma unroll
    for (int m = 0; m < MT; ++m)
        #pragma unroll
        for (int t = 0; t < NT; ++t) acc[m][t] = (v8f){};

    #pragma unroll 2
    for (int k0 = 0; k0 < K; k0 += 4) {
        v2f a[MT];
        #pragma unroll
        for (int m = 0; m < MT; ++m) {
            float a0 = Ap[m * 16 * LDA];
            float a1 = Ap[m * 16 * LDA + 1];
            if (PRE) {
                a0 *= silu_f(A2p[m * 16 * LDA2]);
                a1 *= silu_f(A2p[m * 16 * LDA2 + 1]);
            }
            a[m].x = a0; a[m].y = a1;
        }
        Ap += 4;
        if (PRE) A2p += 4;

        #pragma unroll
        for (int t = 0; t < NT; ++t) {
            v2f b;
            b.x = Bp[t * 16];            // row k   (immediate offset)
            b.y = Bp[LDB + t * 16];      // row k+1 (immediate offset)
            #pragma unroll
            for (int m = 0; m < MT; ++m)
                acc[m][t] = __builtin_amdgcn_wmma_f32_16x16x4_f32(
                    false, a[m], false, b, (short)0, acc[m][t], false, false);
        }

        Bp += (size_t)4 * LDB;
    }

    #pragma unroll
    for (int m = 0; m < MT; ++m) {
        const int rbase = row0 - l + m * 16 + half * 8;
        #pragma unroll
        for (int t = 0; t < NT; ++t) {
            const int col = col0 + t * 16;
            float bcol = POST ? bias[col] : 0.0f;
            #pragma unroll
            for (int r = 0; r < 8; ++r) {
                float v = acc[m][t][r];
                if (POST) v = softplus_f(v + bcol);
                C[(size_t)(rbase + r) * LDC + col] = v;
            }
        }
    }
}

// ---------------------------------------------------------------------------
// Depthwise causal conv (K=4) + SiLU:  u[t,d] = silu(b[d] + sum_j xc[t-3+j,d]*w[d,j])
// xc = first DINNER columns of xz (row stride 2*DINNER).
// ---------------------------------------------------------------------------
__global__ __launch_bounds__(256)
void conv_silu_kernel(const float* __restrict__ xz,
                      const float* __restrict__ conv_w,
                      const float* __restrict__ conv_b,
                      float* __restrict__ u)
{
    int idx = blockIdx.x * blockDim.x + threadIdx.x;
    if (idx >= L_SEQ * DINNER) return;
    const int t = idx / DINNER;
    const int d = idx % DINNER;

    float acc = conv_b[d];
    #pragma unroll
    for (int j = 0; j < 4; ++j) {
        const int tt = t - 3 + j;
        if (tt >= 0)
            acc = fmaf(xz[(size_t)tt * (2 * DINNER) + d], conv_w[d * 4 + j], acc);
    }
    u[idx] = silu_f(acc);
}

// ---------------------------------------------------------------------------
// Selective scan. One lane owns one channel d with 16 f32 states in VGPRs.
// B_t / C_t (16+16 floats, shared by all channels) are chunk-staged in LDS.
// y[t,d] = sum_n h_n * C_t[n] + D[d]*u[t,d];  h_n = exp(delta*A_n)*h_n + delta*u*B_t[n]
// y aliases the u buffer (strictly element-wise read-then-write).
// ---------------------------------------------------------------------------
#define TCHUNK 64

__global__ __launch_bounds__(256)
void scan_kernel(const float* __restrict__ u,
                 const float* __restrict__ delta,
                 const float* __restrict__ xdbl,   // L x 96, B at col 64, C at col 80
                 const float* __restrict__ A_log,
                 const float* __restrict__ D_param,
                 float* __restrict__ y)
{
    __shared__ float sB[TCHUNK][DSTATE];
    __shared__ float sC[TCHUNK][DSTATE];

    const int d = blockIdx.x * blockDim.x + threadIdx.x;   // 8 blocks x 256 = 2048

    float a[DSTATE], h[DSTATE];
    #pragma unroll
    for (int n = 0; n < DSTATE; ++n) {
        a[n] = -__expf(A_log[d * DSTATE + n]);
        h[n] = 0.0f;
    }
    const float Dd = D_param[d];

    for (int c0 = 0; c0 < L_SEQ; c0 += TCHUNK) {
        __syncthreads();
        for (int i = threadIdx.x; i < TCHUNK * 32; i += 256) {
            const int t = i >> 5, j = i & 31;
            const float v = xdbl[(size_t)(c0 + t) * XDBL_W + DTRANK + j];
            if (j < DSTATE) sB[t][j] = v;
            else            sC[t][j - DSTATE] = v;
        }
        __syncthreads();

        for (int t = 0; t < TCHUNK; ++t) {
            const int tt = c0 + t;
            const float ut = u[(size_t)tt * DINNER + d];
            const float dt = delta[(size_t)tt * DINNER + d];
            const float du = dt * ut;
            float acc = Dd * ut;
            #pragma unroll
            for (int n = 0; n < DSTATE; ++n) {
                h[n] = fmaf(__expf(dt * a[n]), h[n], du * sB[t][n]);
                acc  = fmaf(h[n], sC[t][n], acc);
            }
            y[(size_t)tt * DINNER + d] = acc;
        }
    }
}

// ---------------------------------------------------------------------------
// Host-side launch helper
// ---------------------------------------------------------------------------
template <int PRE, int POST, int M, int N, int K,
          int LDA, int LDA2, int LDB, int LDC, int MT, int NT>
static void launch_gemm(const float* A, const float* A2, const float* B,
                        const float* bias, float* C, hipStream_t s)
{
    constexpr int waves  = ((M >> 4) / MT) * ((N >> 4) / NT);
    constexpr int blocks = (waves + 3) / 4;        // 128 threads = 4 waves/block
    gemm_wmma_f32<PRE, POST, M, N, K, LDA, LDA2, LDB, LDC, MT, NT>
        <<<blocks, 128, 0, s>>>(A, A2, B, bias, C);
}

extern "C" void kernel_launch(void* const* d_in, const int* in_sizes, int n_in,
                              void* d_out, int out_size, void* d_ws, size_t ws_size,
                              hipStream_t stream)
{
    const float* x      = (const float*)d_in[0];
    const float* W_in   = (const float*)d_in[1];
    const float* conv_w = (const float*)d_in[2];
    const float* conv_b = (const float*)d_in[3];
    const float* W_x    = (const float*)d_in[4];
    const float* W_dt   = (const float*)d_in[5];
    const float* b_dt   = (const float*)d_in[6];
    const float* A_log  = (const float*)d_in[7];
    const float* D_par  = (const float*)d_in[8];
    const float* W_out  = (const float*)d_in[9];
    float* out = (float*)d_out;

    // workspace layout (floats)
    float* ws    = (float*)d_ws;
    float* xz    = ws;                                    // L x 2*DINNER  (32 MB)
    float* u     = xz    + (size_t)L_SEQ * 2 * DINNER;    // L x DINNER    (16 MB); y aliases u
    float* xdbl  = u     + (size_t)L_SEQ * DINNER;        // L x 96        (0.75 MB)
    float* delta = xdbl  + (size_t)L_SEQ * XDBL_W;        // L x DINNER    (16 MB)
    (void)ws_size; (void)in_sizes; (void)n_in; (void)out_size;

    // 1) xz = x @ W_in   [2048x1024 @ 1024x4096], 32x128 block per wave
    launch_gemm<0, 0, L_SEQ, 2 * DINNER, DMODEL,
                DMODEL, 0, 2 * DINNER, 2 * DINNER, 2, 8>(x, nullptr, W_in, nullptr,
                                                         xz, stream);

    // 2) u = silu(causal_conv4(xc))
    conv_silu_kernel<<<(L_SEQ * DINNER) / 256, 256, 0, stream>>>(xz, conv_w, conv_b, u);

    // 3) x_dbl = u @ W_x   [2048x2048 @ 2048x96], 32x96 block per wave
    launch_gemm<0, 0, L_SEQ, XDBL_W, DINNER,
                DINNER, 0, XDBL_W, XDBL_W, 2, 6>(u, nullptr, W_x, nullptr, xdbl, stream);

    // 4) delta = softplus(x_dbl[:, :64] @ W_dt + b_dt)   [2048x64 @ 64x2048]
    launch_gemm<0, 1, L_SEQ, DINNER, DTRANK,
                XDBL_W, 0, DINNER, DINNER, 2, 8>(xdbl, nullptr, W_dt, b_dt,
                                                 delta, stream);

    // 5) selective scan (y written in-place over u)
    scan_kernel<<<DINNER / 256, 256, 0, stream>>>(u, delta, xdbl, A_log, D_par, u);

    // 6) out = (y * silu(z)) @ W_out   [2048x2048 @ 2048x1024], z = xz[:, DINNER:]
    launch_gemm<1, 0, L_SEQ, DMODEL, DINNER,
                DINNER, 2 * DINNER, DMODEL, DMODEL, 2, 8>(u, xz + DINNER, W_out, nullptr,
                                                          out, stream);
}